// Camera2World_78993038508422
// MI455X (gfx1250) — compile-verified
//
#include <hip/hip_runtime.h>
#include <hip/hip_bf16.h>

// Camera2World: xyz[b,n,c,h,w] = sum_j p2p[b,n,c,j] * [u*d, v*d, d, 1][j], c=0..2
// Shapes fixed by the reference: B=4, N=6, H=512, W=960 (fp32 in/out).
//
// Bandwidth-bound (1.3 FLOP/byte). CDNA5 plan:
//  - depth read stream: Tensor Data Mover (tensor_load_to_lds) DMAs 16 KB
//    tiles into LDS, double-buffered, tracked with TENSORcnt -> deep async
//    prefetch without burning VGPRs on in-flight loads.
//  - matrix: uniform per (b,n) -> scalar loads into SGPRs.
//  - output: three planar streams, full-wave non-temporal b128 stores.
//  - WMMA intentionally unused: the 3x4 uniform matrix on 16x16x4 WMMA wastes
//    13/16 rows and forces half-empty-lane result stores, reducing effective
//    store bandwidth on a memory-roofline kernel.

typedef __attribute__((ext_vector_type(4))) float v4f;
typedef __attribute__((ext_vector_type(4))) unsigned int u32x4;
typedef __attribute__((ext_vector_type(8))) int i32x8;
typedef __attribute__((ext_vector_type(4))) int i32x4;

namespace {
constexpr int kB  = 4;
constexpr int kN  = 6;
constexpr int kH  = 512;
constexpr int kW  = 960;
constexpr int kHW = kH * kW;              // 491520 px per (b,n)
constexpr int kW4 = kW / 4;               // 240 float4 groups per image row
constexpr int kThreads = 256;             // 8 wave32 waves
constexpr int kTile = 4096;               // floats per TDM tile (16 KB)
constexpr int kTileGroups = kTile / 4;    // 1024 float4 groups per tile
constexpr int kGroupsPerThread = kTileGroups / kThreads;   // 4
constexpr int kTilesPerBN = kHW / kTile;  // 120
constexpr int kTilesPerBlock = 5;         // pipeline depth per block
constexpr int kBlocksX = kTilesPerBN / kTilesPerBlock;     // 24 (exact)
static_assert(kBlocksX * kTilesPerBlock == kTilesPerBN, "exact tile cover");
static_assert(kTile <= 0xffff, "tile_dim0 is 16-bit in D#");
}

// Issue one TDM load: tile_elems f32 from gsrc into LDS at byte address lds_byte_off.
// D# built per CDNA5 ISA ch.8 (group0/group1 layouts); groups 2/3 unused (2D).
// This toolchain's builtin takes 6 args (g0, g1, g2, g3, g4, cpol).
__device__ __forceinline__ void tdm_load_tile(const float* gsrc,
                                              unsigned lds_byte_off,
                                              int tile_elems) {
    const unsigned long long ga = (unsigned long long)(uintptr_t)gsrc;
    u32x4 g0;
    g0[0] = 1u;                                   // count=1, user descriptor
    g0[1] = lds_byte_off;                         // lds_addr (bytes)
    g0[2] = (unsigned)(ga & 0xffffffffu);         // global_addr[31:0]
    g0[3] = (unsigned)((ga >> 32) & 0x01ffffffu)  // global_addr[56:32]
          | (2u << 30);                           // type=2 ("image")
    i32x8 g1;
    const unsigned te = (unsigned)tile_elems;
    g1[0] = (int)0x00020000u;                     // data_size=2 (4 B/elem)
    g1[1] = (int)((te & 0xffffu) << 16);          // tensor_dim0[15:0]
    g1[2] = (int)((te >> 16) | (1u << 16));       // tensor_dim0[31:16], tensor_dim1=1
    g1[3] = (int)((te & 0xffffu) << 16);          // tile_dim0 = tile_elems
    g1[4] = 1;                                    // tile_dim1=1, tile_dim2=0
    g1[5] = (int)te;                              // tensor_dim0_stride (lo32)
    g1[6] = 0;
    g1[7] = 0;
    const i32x4 z4 = {0, 0, 0, 0};
    const i32x8 z8 = {0, 0, 0, 0, 0, 0, 0, 0};
    __builtin_amdgcn_tensor_load_to_lds(g0, g1, z4, z4, z8, /*cpol=*/0);
}

__global__ __launch_bounds__(kThreads) void Camera2World_78993038508422_kernel(
    const float* __restrict__ depth,   // [B*N, H*W]
    const float* __restrict__ p2p,     // [B*N, 4, 4]
    float* __restrict__ out)           // [B*N, 3, H*W]
{
    __shared__ float buf[2][kTile];    // 32 KB double buffer

    const int bn = blockIdx.y;

    // Uniform per-block matrix rows -> scalar loads.
    const float* __restrict__ P = p2p + bn * 16;
    const float p00 = P[0], p01 = P[1], p02 = P[2],  p03 = P[3];
    const float p10 = P[4], p11 = P[5], p12 = P[6],  p13 = P[7];
    const float p20 = P[8], p21 = P[9], p22 = P[10], p23 = P[11];

    const float* __restrict__ dbase = depth + (size_t)bn * kHW;
    float* __restrict__ ox = out + (size_t)bn * (3 * (size_t)kHW);
    float* __restrict__ oy = ox + kHW;
    float* __restrict__ oz = oy + kHW;

    const int  tile0   = blockIdx.x * kTilesPerBlock;
    const bool isWave0 = (threadIdx.x < 32);
    // LDS byte addresses: flat->LDS mapping keeps only addr[31:0] (ISA 10.2).
    const unsigned lds0 = (unsigned)(uintptr_t)&buf[0][0];
    const unsigned lds1 = (unsigned)(uintptr_t)&buf[1][0];

    // Prologue: DMA tile 0 into buffer 0.
    if (isWave0) {
        tdm_load_tile(dbase + (size_t)tile0 * kTile, lds0, kTile);
    }

    for (int t = 0; t < kTilesPerBlock; ++t) {
        if (isWave0) {
            if (t + 1 < kTilesPerBlock) {
                // Overlap: DMA tile t+1 into the other buffer, then wait for
                // tile t only (TDM ops complete in order; cnt<=1 == t done).
                tdm_load_tile(dbase + (size_t)(tile0 + t + 1) * kTile,
                              ((t + 1) & 1) ? lds1 : lds0, kTile);
                __builtin_amdgcn_s_wait_tensorcnt(1);
            } else {
                __builtin_amdgcn_s_wait_tensorcnt(0);
            }
        }
        __syncthreads();   // tile t visible in LDS to all 8 waves

        const float* __restrict__ lbuf = (t & 1) ? &buf[1][0] : &buf[0][0];
        const int gbase = (tile0 + t) * kTileGroups;   // tile's first group

        #pragma unroll
        for (int k = 0; k < kGroupsPerThread; ++k) {
            const int gl = threadIdx.x + k * kThreads;       // group in tile
            const v4f d  = *reinterpret_cast<const v4f*>(lbuf + 4 * gl);

            const int   g   = gbase + gl;
            const int   v   = g / kW4;               // pixel row
            const int   u0  = (g - v * kW4) * 4;     // first pixel col
            const float fv  = (float)v;
            const float fu0 = (float)u0;

            // a_i = P[i,1]*v + P[i,2] shared by the 4 pixels of the group.
            const float a0 = __builtin_fmaf(p01, fv, p02);
            const float a1 = __builtin_fmaf(p11, fv, p12);
            const float a2 = __builtin_fmaf(p21, fv, p22);

            v4f x, y, z;
            #pragma unroll
            for (int e = 0; e < 4; ++e) {
                const float u  = fu0 + (float)e;
                const float w0 = __builtin_fmaf(p00, u, a0);
                const float w1 = __builtin_fmaf(p10, u, a1);
                const float w2 = __builtin_fmaf(p20, u, a2);
                x[e] = __builtin_fmaf(w0, d[e], p03);
                y[e] = __builtin_fmaf(w1, d[e], p13);
                z[e] = __builtin_fmaf(w2, d[e], p23);
            }

            // Write-once planar streams: non-temporal b128 stores.
            __builtin_nontemporal_store(x, reinterpret_cast<v4f*>(ox + 4 * g));
            __builtin_nontemporal_store(y, reinterpret_cast<v4f*>(oy + 4 * g));
            __builtin_nontemporal_store(z, reinterpret_cast<v4f*>(oz + 4 * g));
        }

        __syncthreads();   // all waves done reading buf[t&1] before it is
                           // re-targeted by the DMA issued next iteration
    }
}

extern "C" void kernel_launch(void* const* d_in, const int* in_sizes, int n_in,
                              void* d_out, int out_size, void* d_ws, size_t ws_size,
                              hipStream_t stream) {
    (void)in_sizes; (void)n_in; (void)out_size; (void)d_ws; (void)ws_size;
    const float* depth = (const float*)d_in[0];  // [4,6,512,960] f32
    const float* p2p   = (const float*)d_in[1];  // [4,6,4,4]     f32
    float* out         = (float*)d_out;          // [4,6,3,512,960] f32

    dim3 block(kThreads);
    dim3 grid(kBlocksX, kB * kN);   // (24, 24) = 576 blocks, exact cover
    Camera2World_78993038508422_kernel<<<grid, block, 0, stream>>>(depth, p2p, out);
}